// RDurendal_74423193305788
// MI455X (gfx1250) — compile-verified
//
#include <hip/hip_runtime.h>
#include <cstddef>

// ---- problem constants (match reference) ----
#define CN   100000
#define CIN  128
#define CH1  128
#define CH2  64
#define CR   4
#define CE   500000
#define CEL  200000

typedef float v2f __attribute__((ext_vector_type(2)));
typedef float v8f __attribute__((ext_vector_type(8)));

__device__ __forceinline__ v8f wmma_k4(v2f a, v2f b, v8f c) {
  // V_WMMA_F32_16X16X4_F32 : D = A(16x4 f32) * B(4x16 f32) + C(16x16 f32)
  return __builtin_amdgcn_wmma_f32_16x16x4_f32(false, a, false, b, (short)0, c,
                                               false, false);
}

__device__ __forceinline__ float sigmoidf_(float x) {
  return 1.0f / (1.0f + __expf(-x));
}

// ---------------------------------------------------------------- zero
__global__ void zero_f32(float* __restrict__ p, size_t n) {
  size_t i = (size_t)blockIdx.x * blockDim.x + threadIdx.x;
  size_t stride = (size_t)gridDim.x * blockDim.x;
  for (; i < n; i += stride) p[i] = 0.0f;
}

// -------------------------------------------- weight fragment packing
// Non-transposed: B used as (K x C);  pk[b][p][c] = {B[2p][c], B[2p+1][c]}
__global__ void pack_b(const float* __restrict__ src, v2f* __restrict__ dst,
                       int K, int C, int batch) {
  int i = blockIdx.x * blockDim.x + threadIdx.x;
  int per = (K / 2) * C;
  if (i >= batch * per) return;
  int b = i / per;
  int rem = i - b * per;
  int p = rem / C, c = rem - p * C;
  const float* s = src + (size_t)b * K * C;
  v2f v;
  v.x = s[(size_t)(2 * p) * C + c];
  v.y = s[(size_t)(2 * p + 1) * C + c];
  dst[i] = v;
}

// Transposed: src is (RW x K) row-major, used as B[k][row] = src[row][k]
// pk[p][row] = {src[row][2p], src[row][2p+1]}
__global__ void pack_bT(const float* __restrict__ src, v2f* __restrict__ dst,
                        int RW, int K) {
  int i = blockIdx.x * blockDim.x + threadIdx.x;
  if (i >= (K / 2) * RW) return;
  int p = i / RW, row = i - p * RW;
  v2f v;
  v.x = src[(size_t)row * K + 2 * p];
  v.y = src[(size_t)row * K + 2 * p + 1];
  dst[i] = v;
}

// ------------------------------------------------- edge gather/scatter
// one wave32 per edge; 128 channels -> float4 per lane; atomic segment-sum
__global__ void scatter_accum(const float* __restrict__ feat,   // (N,128)
                              const int* __restrict__ eidx,     // (R,2,E)
                              float* __restrict__ agg,          // (R,N,128)
                              float* __restrict__ deg)          // (R,N) or null
{
  int wave = (int)(((size_t)blockIdx.x * blockDim.x + threadIdx.x) >> 5);
  int lane = threadIdx.x & 31;
  if (wave >= CR * CE) return;
  int r = wave / CE;
  int e = wave - r * CE;
  int src = eidx[((size_t)r * 2 + 0) * CE + e];
  int dst = eidx[((size_t)r * 2 + 1) * CE + e];
  const float4* frow = (const float4*)(feat + (size_t)src * 128);
  float4 v = frow[lane];
  float* arow = agg + ((size_t)r * CN + dst) * 128 + lane * 4;
  atomicAdd(arow + 0, v.x);
  atomicAdd(arow + 1, v.y);
  atomicAdd(arow + 2, v.z);
  atomicAdd(arow + 3, v.w);
  if (deg && lane == 0) atomicAdd(&deg[(size_t)r * CN + dst], 1.0f);
}

// ------------------------------------------------- conv GEMM (WMMA f32)
// out[r,row,col] = sum_k (agg[r,row,k]/max(deg,1)) * W[r,k,col] + bias[r,col]
// blockDim = 256 (8 waves).  Weights pre-packed into v2f fragments.
template <int ROWS, int NOUT>
__global__ void conv_gemm(const float* __restrict__ aggin,
                          const float* __restrict__ deg,
                          const float* __restrict__ Wpk,    // (R, 64, NOUT) v2f
                          const float* __restrict__ bias,   // (R,NOUT)
                          float* __restrict__ outp, int outStride)
{
  __shared__ float As[ROWS][132];
  const int row0 = blockIdx.x * ROWS;
  const int r = blockIdx.y;
  for (int idx = threadIdx.x; idx < ROWS * 128; idx += blockDim.x) {
    int rr = idx >> 7, k = idx & 127;
    float d = deg[(size_t)r * CN + row0 + rr];
    float inv = 1.0f / fmaxf(d, 1.0f);
    As[rr][k] = aggin[((size_t)r * CN + row0 + rr) * 128 + k] * inv;
  }
  __syncthreads();

  const int lane = threadIdx.x & 31;
  const int w = threadIdx.x >> 5;
  const int NCT = NOUT / 16;
  const int sub = w / NCT;          // node sub-tile
  const int ct = w % NCT;           // 16-col tile
  const int m = lane & 15;
  const int half = lane >> 4;
  const int col = ct * 16 + m;
  const v2f* Bpk = (const v2f*)Wpk + (size_t)r * 64 * NOUT;

  v8f acc = {};
  for (int kk = 0; kk < 32; ++kk) {
    int p = 2 * kk + half;
    if ((kk & 7) == 0)
      __builtin_prefetch(Bpk + (size_t)(p + 16) * NOUT + col, 0, 1);
    v2f a;
    a.x = As[sub * 16 + m][4 * kk + 2 * half];
    a.y = As[sub * 16 + m][4 * kk + 2 * half + 1];
    v2f b = Bpk[(size_t)p * NOUT + col];
    acc = wmma_k4(a, b, acc);
  }
  float bb = bias[r * NOUT + col];
#pragma unroll
  for (int v = 0; v < 8; ++v) {
    int row = sub * 16 + v + 8 * half;
    outp[((size_t)r * CN + row0 + row) * outStride + col] = acc[v] + bb;
  }
}

// ------------------------------------------------- fused GRU (WMMA f32)
// gi = x @ wih^T + bih ; gh = h @ whh^T + bhh ; gates -> cur (written to d_out)
// wih/whh pre-packed transposed: WI[p][g*H+j] = {wih[g*H+j][2p], wih[g*H+j][2p+1]}
template <int ROWS, int H>
__global__ void gru_fused(const float* __restrict__ xin, int xinStride,
                          const float* __restrict__ hpast,   // (R,N,H)
                          const float* __restrict__ wihpk,   // (H/2, 3H) v2f
                          const float* __restrict__ whhpk,   // (H/2, 3H) v2f
                          const float* __restrict__ bih,
                          const float* __restrict__ bhh,
                          const int* __restrict__ snap,
                          float* __restrict__ curout)
{
  __shared__ float Xs[ROWS][H + 4];
  __shared__ float Hs[ROWS][H + 4];
  const int row0 = blockIdx.x * ROWS;
  const int r = blockIdx.y;
  for (int idx = threadIdx.x; idx < ROWS * H; idx += blockDim.x) {
    int rr = idx / H, k = idx % H;
    Xs[rr][k] = xin[((size_t)r * CN + row0 + rr) * xinStride + k];
    Hs[rr][k] = hpast[((size_t)r * CN + row0 + rr) * H + k];
  }
  __syncthreads();

  const int lane = threadIdx.x & 31;
  const int w = threadIdx.x >> 5;
  const int NCT = H / 16;
  const int sub = w / NCT;
  const int ct = w % NCT;
  const int m = lane & 15;
  const int half = lane >> 4;
  const int j = ct * 16 + m;   // output gate column within H
  const v2f* WI = (const v2f*)wihpk;
  const v2f* WH = (const v2f*)whhpk;

  v8f air = {}, aiz = {}, ain = {}, ahr = {}, ahz = {}, ahn = {};
  for (int kk = 0; kk < H / 4; ++kk) {
    int k0 = 4 * kk + 2 * half;
    int p = 2 * kk + half;
    v2f ax, ah;
    ax.x = Xs[sub * 16 + m][k0];
    ax.y = Xs[sub * 16 + m][k0 + 1];
    ah.x = Hs[sub * 16 + m][k0];
    ah.y = Hs[sub * 16 + m][k0 + 1];
    const v2f* wi = WI + (size_t)p * (3 * H) + j;
    const v2f* wh = WH + (size_t)p * (3 * H) + j;
    air = wmma_k4(ax, wi[0 * H], air);
    ahr = wmma_k4(ah, wh[0 * H], ahr);
    aiz = wmma_k4(ax, wi[1 * H], aiz);
    ahz = wmma_k4(ah, wh[1 * H], ahz);
    ain = wmma_k4(ax, wi[2 * H], ain);
    ahn = wmma_k4(ah, wh[2 * H], ahn);
  }

  const int sn = *snap;
  const float bir = bih[j], biz = bih[H + j], bin_ = bih[2 * H + j];
  const float bhr = bhh[j], bhz = bhh[H + j], bhn = bhh[2 * H + j];
#pragma unroll
  for (int v = 0; v < 8; ++v) {
    int row = sub * 16 + v + 8 * half;
    float ir = air[v] + bir, iz = aiz[v] + biz, in_ = ain[v] + bin_;
    float hr = ahr[v] + bhr, hz = ahz[v] + bhz, hn = ahn[v] + bhn;
    float rg = sigmoidf_(ir + hr);
    float zg = sigmoidf_(iz + hz);
    float ng = tanhf(in_ + rg * hn);
    float hv = Hs[row][j];
    float cur = (1.0f - zg) * ng + zg * hv;
    if (sn == 0) cur = Xs[row][j];
    curout[((size_t)r * CN + row0 + row) * H + j] = cur;
  }
}

// ------------------------------- semantic attention score (WMMA + reduce)
// wsum[r] += sum_{nodes,j} tanh(cur @ aw + ab)[n,j] * aq[j]
template <int ROWS, int H>
__global__ void att_score(const float* __restrict__ cur,   // (R,N,H)
                          const float* __restrict__ awpk,  // (H/2, H) v2f
                          const float* __restrict__ ab,
                          const float* __restrict__ aq,
                          float* __restrict__ wsum)
{
  __shared__ float Cs[ROWS][H + 4];
  const int row0 = blockIdx.x * ROWS;
  const int r = blockIdx.y;
  for (int idx = threadIdx.x; idx < ROWS * H; idx += blockDim.x) {
    int rr = idx / H, k = idx % H;
    Cs[rr][k] = cur[((size_t)r * CN + row0 + rr) * H + k];
  }
  __syncthreads();

  const int lane = threadIdx.x & 31;
  const int w = threadIdx.x >> 5;
  const int NCT = H / 16;
  const int sub = w / NCT;
  const int ct = w % NCT;
  const int m = lane & 15;
  const int half = lane >> 4;
  const int col = ct * 16 + m;
  const v2f* Bpk = (const v2f*)awpk;

  v8f acc = {};
  for (int kk = 0; kk < H / 4; ++kk) {
    int p = 2 * kk + half;
    v2f a;
    a.x = Cs[sub * 16 + m][4 * kk + 2 * half];
    a.y = Cs[sub * 16 + m][4 * kk + 2 * half + 1];
    v2f b = Bpk[(size_t)p * H + col];
    acc = wmma_k4(a, b, acc);
  }
  float abj = ab[col], qj = aq[col];
  float s = 0.0f;
#pragma unroll
  for (int v = 0; v < 8; ++v) s += tanhf(acc[v] + abj) * qj;
#pragma unroll
  for (int off = 16; off > 0; off >>= 1) s += __shfl_xor(s, off, 32);
  if (lane == 0) atomicAdd(&wsum[r], s);
}

// ---------------------------------------------------------- softmax(R=4)
__global__ void softmax_beta(const float* __restrict__ wsum,
                             float* __restrict__ beta, float invN) {
  if (blockIdx.x == 0 && threadIdx.x == 0) {
    float w[CR];
    float mx = -1e30f;
    for (int i = 0; i < CR; ++i) { w[i] = wsum[i] * invN; mx = fmaxf(mx, w[i]); }
    float s = 0.0f;
    for (int i = 0; i < CR; ++i) { w[i] = __expf(w[i] - mx); s += w[i]; }
    for (int i = 0; i < CR; ++i) beta[i] = w[i] / s;
  }
}

// ----------------------------------------- z1 = sum_r beta[r] * cur1[r]
__global__ void combine_z1(const float* __restrict__ cur1,   // (R,N,128)
                           const float* __restrict__ beta,
                           float* __restrict__ z1)           // (N,128)
{
  int i = blockIdx.x * blockDim.x + threadIdx.x;
  if (i >= CN * 128) return;
  float acc = beta[0] * cur1[i];
  acc += beta[1] * cur1[(size_t)1 * CN * 128 + i];
  acc += beta[2] * cur1[(size_t)2 * CN * 128 + i];
  acc += beta[3] * cur1[(size_t)3 * CN * 128 + i];
  z1[i] = acc;
}

// -------------------- z2 combine + post-linear (64 -> 2) : one wave/node
__global__ void z2_nodeout(const float* __restrict__ cur2,  // (R,N,64)
                           const float* __restrict__ beta,
                           const float* __restrict__ post_w, // (64,2)
                           const float* __restrict__ post_b, // (2,)
                           float* __restrict__ nodeout)      // (N,2)
{
  int wave = (int)(((size_t)blockIdx.x * blockDim.x + threadIdx.x) >> 5);
  int lane = threadIdx.x & 31;
  if (wave >= CN) return;
  int c0 = 2 * lane, c1 = c0 + 1;
  size_t base = (size_t)wave * 64;
  float z0 = 0.0f, z1v = 0.0f;
#pragma unroll
  for (int r = 0; r < CR; ++r) {
    float br = beta[r];
    z0 += br * cur2[(size_t)r * CN * 64 + base + c0];
    z1v += br * cur2[(size_t)r * CN * 64 + base + c1];
  }
  float p0 = z0 * post_w[c0 * 2 + 0] + z1v * post_w[c1 * 2 + 0];
  float p1 = z0 * post_w[c0 * 2 + 1] + z1v * post_w[c1 * 2 + 1];
#pragma unroll
  for (int off = 16; off > 0; off >>= 1) {
    p0 += __shfl_xor(p0, off, 32);
    p1 += __shfl_xor(p1, off, 32);
  }
  if (lane == 0) {
    nodeout[(size_t)wave * 2 + 0] = p0 + post_b[0];
    nodeout[(size_t)wave * 2 + 1] = p1 + post_b[1];
  }
}

// ------------------------------------------------------ edge scoring
__global__ void edge_score(const float* __restrict__ nodeout,  // (N,2)
                           const int* __restrict__ eli,        // (R,2,EL)
                           const float* __restrict__ rel_emb,  // (R,2)
                           float* __restrict__ scores)         // (R,EL)
{
  int i = blockIdx.x * blockDim.x + threadIdx.x;
  if (i >= CR * CEL) return;
  int r = i / CEL, e = i - r * CEL;
  int h = eli[((size_t)r * 2 + 0) * CEL + e];
  int t = eli[((size_t)r * 2 + 1) * CEL + e];
  float hr = nodeout[(size_t)h * 2 + 0], hi = nodeout[(size_t)h * 2 + 1];
  float tr = nodeout[(size_t)t * 2 + 0], ti = nodeout[(size_t)t * 2 + 1];
  float rr = rel_emb[r * 2 + 0], ri = rel_emb[r * 2 + 1];
  scores[i] = rr * (hr * tr + hi * ti) + ri * (hr * ti - hi * tr);
}

// =====================================================================
extern "C" void kernel_launch(void* const* d_in, const int* in_sizes, int n_in,
                              void* d_out, int out_size, void* d_ws,
                              size_t ws_size, hipStream_t stream) {
  (void)in_sizes; (void)n_in; (void)out_size; (void)ws_size;
  const float* x       = (const float*)d_in[0];
  const int* eindex    = (const int*)d_in[1];
  const int* eli       = (const int*)d_in[2];
  const float* past1   = (const float*)d_in[3];
  const float* past2   = (const float*)d_in[4];
  const float* W1      = (const float*)d_in[5];
  const float* b1      = (const float*)d_in[6];
  const float* W2      = (const float*)d_in[7];
  const float* b2      = (const float*)d_in[8];
  const float* g1wih   = (const float*)d_in[9];
  const float* g1whh   = (const float*)d_in[10];
  const float* g1bih   = (const float*)d_in[11];
  const float* g1bhh   = (const float*)d_in[12];
  const float* g2wih   = (const float*)d_in[13];
  const float* g2whh   = (const float*)d_in[14];
  const float* g2bih   = (const float*)d_in[15];
  const float* g2bhh   = (const float*)d_in[16];
  const float* aw1     = (const float*)d_in[17];
  const float* ab1     = (const float*)d_in[18];
  const float* aq1     = (const float*)d_in[19];
  const float* aw2     = (const float*)d_in[20];
  const float* ab2     = (const float*)d_in[21];
  const float* aq2     = (const float*)d_in[22];
  const float* post_w  = (const float*)d_in[23];
  const float* post_b  = (const float*)d_in[24];
  const float* rel_emb = (const float*)d_in[25];
  const int* snap      = (const int*)d_in[26];

  // workspace layout (floats)
  float* ws = (float*)d_ws;
  float* agg     = ws;                                  // R*N*128
  float* deg     = agg + (size_t)CR * CN * 128;         // R*N
  float* wsum    = deg + (size_t)CR * CN;               // 8 (w1sum, w2sum)
  float* beta    = wsum + 8;                            // 8 (beta1, beta2)
  float* z1      = beta + 8;                            // N*128
  float* nodeout = z1 + (size_t)CN * 128;               // N*2
  float* w1pk    = nodeout + (size_t)CN * 2;            // R*64*128*2
  float* w2pk    = w1pk + (size_t)CR * 64 * 128 * 2;    // R*64*64*2
  float* g1wihpk = w2pk + (size_t)CR * 64 * 64 * 2;     // 64*384*2
  float* g1whhpk = g1wihpk + (size_t)64 * 384 * 2;
  float* g2wihpk = g1whhpk + (size_t)64 * 384 * 2;      // 32*192*2
  float* g2whhpk = g2wihpk + (size_t)32 * 192 * 2;
  float* aw1pk   = g2whhpk + (size_t)32 * 192 * 2;      // 64*128*2
  float* aw2pk   = aw1pk + (size_t)64 * 128 * 2;        // 32*64*2

  // output layout: scores (R,EL) | cur1 (R,N,128) | cur2 (R,N,64)
  float* scores = (float*)d_out;
  float* cur1   = scores + (size_t)CR * CEL;
  float* cur2   = cur1 + (size_t)CR * CN * 128;

  const int edge_waves = CR * CE;                       // one wave per edge
  const int edge_blocks = (edge_waves * 32 + 255) / 256;

  // ---- one-time weight fragment packing (tiny; runs every launch) ----
  pack_b<<<(CR * 64 * 128 + 255) / 256, 256, 0, stream>>>(W1, (v2f*)w1pk, 128, 128, CR);
  pack_b<<<(CR * 64 * 64 + 255) / 256, 256, 0, stream>>>(W2, (v2f*)w2pk, 128, 64, CR);
  pack_bT<<<(64 * 384 + 255) / 256, 256, 0, stream>>>(g1wih, (v2f*)g1wihpk, 384, 128);
  pack_bT<<<(64 * 384 + 255) / 256, 256, 0, stream>>>(g1whh, (v2f*)g1whhpk, 384, 128);
  pack_bT<<<(32 * 192 + 255) / 256, 256, 0, stream>>>(g2wih, (v2f*)g2wihpk, 192, 64);
  pack_bT<<<(32 * 192 + 255) / 256, 256, 0, stream>>>(g2whh, (v2f*)g2whhpk, 192, 64);
  pack_b<<<(64 * 128 + 255) / 256, 256, 0, stream>>>(aw1, (v2f*)aw1pk, 128, 128, 1);
  pack_b<<<(32 * 64 + 255) / 256, 256, 0, stream>>>(aw2, (v2f*)aw2pk, 64, 64, 1);

  // ---- layer 1 ----
  zero_f32<<<4096, 256, 0, stream>>>(agg, (size_t)CR * CN * 128 + CR * CN + 16);
  scatter_accum<<<edge_blocks, 256, 0, stream>>>(x, eindex, agg, deg);
  dim3 g1(CN / 16, CR);
  conv_gemm<16, 128><<<g1, 256, 0, stream>>>(agg, deg, w1pk, b1, agg, 128);
  gru_fused<16, 128><<<g1, 256, 0, stream>>>(agg, 128, past1, g1wihpk, g1whhpk,
                                             g1bih, g1bhh, snap, cur1);
  att_score<16, 128><<<g1, 256, 0, stream>>>(cur1, aw1pk, ab1, aq1, wsum);
  softmax_beta<<<1, 32, 0, stream>>>(wsum, beta, 1.0f / (float)CN);
  combine_z1<<<(CN * 128) / 256, 256, 0, stream>>>(cur1, beta, z1);

  // ---- layer 2 ----
  zero_f32<<<4096, 256, 0, stream>>>(agg, (size_t)CR * CN * 128);
  scatter_accum<<<edge_blocks, 256, 0, stream>>>(z1, eindex, agg, nullptr);
  dim3 g2(CN / 32, CR);
  // writes 64 output cols into cols [0,64) of agg rows (stride 128, in place)
  conv_gemm<32, 64><<<g2, 256, 0, stream>>>(agg, deg, w2pk, b2, agg, 128);
  gru_fused<32, 64><<<g2, 256, 0, stream>>>(agg, 128, past2, g2wihpk, g2whhpk,
                                            g2bih, g2bhh, snap, cur2);
  att_score<32, 64><<<g2, 256, 0, stream>>>(cur2, aw2pk, ab2, aq2, wsum + 4);
  softmax_beta<<<1, 32, 0, stream>>>(wsum + 4, beta + 4, 1.0f / (float)CN);

  // ---- head ----
  z2_nodeout<<<CN / 8, 256, 0, stream>>>(cur2, beta + 4, post_w, post_b, nodeout);
  edge_score<<<(CR * CEL) / 256, 256, 0, stream>>>(nodeout, eli, rel_emb, scores);
}